// CrossTransformer_14886356648888
// MI455X (gfx1250) — compile-verified
//
#include <hip/hip_runtime.h>

typedef float v2f __attribute__((ext_vector_type(2)));
typedef float v8f __attribute__((ext_vector_type(8)));

#define EPS 1e-5f
#define SQRT_C 22.627416997969522f   // sqrt(512)

// ---------------- dimensions ----------------
// B=64 S=31 CTX=2048 C=512 H=8 DH=64
constexpr int    Bn   = 64;
constexpr int    Sn   = 31;
constexpr int    CTXn = 2048;
constexpr int    Cn   = 512;
constexpr size_t NQ   = (size_t)Bn * Sn;        // 1984 rows (31*64)
constexpr size_t NCTX = (size_t)Bn * CTXn;      // 131072 rows
constexpr size_t BH   = (size_t)Bn * 8;         // 512
constexpr size_t SS   = (size_t)Sn * Sn;        // 961

// ---------------- workspace layout (floats) ----------------
constexpr size_t OFF_MU1  = 0;
constexpr size_t OFF_VAR1 = OFF_MU1 + Cn;
constexpr size_t OFF_MU2  = OFF_VAR1 + Cn;
constexpr size_t OFF_VAR2 = OFF_MU2 + SS;
constexpr size_t OFF_XN   = 4096;
constexpr size_t OFF_Q    = OFF_XN  + NQ * Cn;
constexpr size_t OFF_K    = OFF_Q   + NQ * Cn;
constexpr size_t OFF_V    = OFF_K   + NQ * Cn;
constexpr size_t OFF_EN   = OFF_V   + NQ * Cn;
constexpr size_t OFF_ATT  = OFF_EN  + BH * SS;
constexpr size_t OFF_O    = OFF_ATT + BH * SS;
constexpr size_t OFF_XSA  = OFF_O   + NQ * Cn;
constexpr size_t OFF_QN   = OFF_XSA + NQ * Cn;
constexpr size_t OFF_Q2   = OFF_QN  + NQ * Cn;
constexpr size_t OFF_O2   = OFF_Q2  + NQ * Cn;
constexpr size_t OFF_XCA  = OFF_O2  + NQ * Cn;
constexpr size_t OFF_H1   = OFF_XCA + NQ * Cn;
constexpr size_t OFF_H1N  = OFF_H1  + NQ * 2 * Cn;
constexpr size_t OFF_K2   = OFF_H1N + NQ * 2 * Cn;
constexpr size_t OFF_V2   = OFF_K2  + NCTX * Cn;
constexpr size_t OFF_A2   = OFF_K2;   // attn2 reuses K2 region (K2 dead after energy2)

#define WMMA_F32(a, b, c) \
  __builtin_amdgcn_wmma_f32_16x16x4_f32(false, (a), false, (b), (short)0, (c), false, false)

// ============================================================
// Generic GEMM: C[m][n] = sum_k A[m][k]*W[n][k] (+bias[n]) (+res[m][n])
// A: MxK row-major, W: NxK row-major (torch Linear layout), C: MxN.
// grid = (N/64, M/64), block = 128: 4 waves stride M; each wave owns a
// 16x64 wave tile (4 N-subtiles share one A fragment -> 1.25 loads/WMMA).
// Software double-buffered: iteration k+1's 5 fragment loads issue before
// iteration k's 4 WMMAs, so WMMAs never wait on the newest loads.
// Requires K >= 8 and K % 4 == 0 (K is 512 or 1024 here).
// ============================================================
__global__ void __launch_bounds__(128)
gemm_xwt(const float* __restrict__ A, const float* __restrict__ W,
         const float* __restrict__ bias, const float* __restrict__ res,
         float* __restrict__ Cout, int M, int N, int K) {
  const int w    = threadIdx.x >> 5;
  const int lane = threadIdx.x & 31;
  const int lm = lane & 15;
  const int ko = (lane >> 4) * 2;
  const int n0 = blockIdx.x * 64;
  const int m0 = blockIdx.y * 64 + w * 16;
  const float* ap = A + (size_t)(m0 + lm) * K + ko;
  const float* w0 = W + (size_t)(n0 + lm) * K + ko;
  const float* w1 = w0 + (size_t)16 * K;
  const float* w2 = w0 + (size_t)32 * K;
  const float* w3 = w0 + (size_t)48 * K;
  v8f acc[4] = {};
  v2f a  = *(const v2f*)(ap);
  v2f b0 = *(const v2f*)(w0);
  v2f b1 = *(const v2f*)(w1);
  v2f b2 = *(const v2f*)(w2);
  v2f b3 = *(const v2f*)(w3);
#pragma unroll 2
  for (int kk = 4; kk < K; kk += 4) {
    v2f an  = *(const v2f*)(ap + kk);
    v2f b0n = *(const v2f*)(w0 + kk);
    v2f b1n = *(const v2f*)(w1 + kk);
    v2f b2n = *(const v2f*)(w2 + kk);
    v2f b3n = *(const v2f*)(w3 + kk);
    acc[0] = WMMA_F32(a, b0, acc[0]);
    acc[1] = WMMA_F32(a, b1, acc[1]);
    acc[2] = WMMA_F32(a, b2, acc[2]);
    acc[3] = WMMA_F32(a, b3, acc[3]);
    a = an; b0 = b0n; b1 = b1n; b2 = b2n; b3 = b3n;
  }
  acc[0] = WMMA_F32(a, b0, acc[0]);
  acc[1] = WMMA_F32(a, b1, acc[1]);
  acc[2] = WMMA_F32(a, b2, acc[2]);
  acc[3] = WMMA_F32(a, b3, acc[3]);
#pragma unroll
  for (int j = 0; j < 4; ++j) {
#pragma unroll
    for (int r = 0; r < 8; ++r) {
      int m = m0 + r + (lane >> 4) * 8;
      int n = n0 + j * 16 + lm;
      float v = acc[j][r];
      if (bias) v += bias[n];
      if (res)  v += res[(size_t)m * N + n];
      Cout[(size_t)m * N + n] = v;
    }
  }
}

// ============================================================
// Per-channel batch stats: mu/var over `rows` samples, channel stride C.
// grid = C channels, block = 256.
// ============================================================
__global__ void bn_stats_ch(const float* __restrict__ x, int rows, int C,
                            float* __restrict__ mu, float* __restrict__ var) {
  const int c = blockIdx.x;
  __shared__ float s1[256], s2[256];
  float sum = 0.f, sq = 0.f;
  for (int r = threadIdx.x; r < rows; r += 256) {
    float v = x[(size_t)r * C + c];
    sum += v; sq += v * v;
  }
  s1[threadIdx.x] = sum; s2[threadIdx.x] = sq;
  __syncthreads();
  for (int s = 128; s > 0; s >>= 1) {
    if (threadIdx.x < s) { s1[threadIdx.x] += s1[threadIdx.x + s];
                           s2[threadIdx.x] += s2[threadIdx.x + s]; }
    __syncthreads();
  }
  if (threadIdx.x == 0) {
    float m = s1[0] / rows;
    mu[c] = m;
    var[c] = s2[0] / rows - m * m;
  }
}

__global__ void bn_apply(const float* __restrict__ x, const float* __restrict__ mu,
                         const float* __restrict__ var, const float* __restrict__ g,
                         const float* __restrict__ bta, float* __restrict__ y,
                         int n, int C) {
  int i = blockIdx.x * 256 + threadIdx.x;
  if (i >= n) return;
  int c = i % C;
  y[i] = (x[i] - mu[c]) * rsqrtf(var[c] + EPS) * g[c] + bta[c];
}

// ============================================================
// Self-attention energy: E[bh,q,k] = sum_dh Q[b,q,h,dh]*K[b,k,h,dh]
// 31x31 padded to 32x32 (zero fragments). grid = BH, block = 128
// (4 waves: 2x2 tile grid).
// ============================================================
__global__ void self_energy(const float* __restrict__ Q, const float* __restrict__ K,
                            float* __restrict__ E) {
  const int bh = blockIdx.x, b = bh >> 3, h = bh & 7;
  const int w = threadIdx.x >> 5, lane = threadIdx.x & 31;
  const int mt = w >> 1, nt = w & 1;
  const int lm = lane & 15, ko = (lane >> 4) * 2;
  const float* qp = Q + (size_t)b * Sn * Cn + h * 64;
  const float* kp = K + (size_t)b * Sn * Cn + h * 64;
  const int mr = mt * 16 + lm, nr = nt * 16 + lm;
  v8f acc = {};
#pragma unroll
  for (int kk = 0; kk < 64; kk += 4) {
    v2f a, bf;
    if (mr < Sn) a = *(const v2f*)(qp + (size_t)mr * Cn + kk + ko);
    else { a.x = 0.f; a.y = 0.f; }
    if (nr < Sn) bf = *(const v2f*)(kp + (size_t)nr * Cn + kk + ko);
    else { bf.x = 0.f; bf.y = 0.f; }
    acc = WMMA_F32(a, bf, acc);
  }
#pragma unroll
  for (int r = 0; r < 8; ++r) {
    int m = mt * 16 + r + (lane >> 4) * 8;
    int n = nt * 16 + lm;
    if (m < Sn && n < Sn) E[(size_t)bh * SS + (size_t)m * Sn + n] = acc[r];
  }
}

// BN(961-ch) + /sqrt(C) + softmax over k (31). grid = BH*S rows, block = 32.
__global__ void softmax_self(const float* __restrict__ E, const float* __restrict__ mu2,
                             const float* __restrict__ var2, const float* __restrict__ g2,
                             const float* __restrict__ b2, float* __restrict__ ATT) {
  const int row = blockIdx.x;
  const int q = row % Sn, bh = row / Sn;
  const int k = threadIdx.x;
  const bool valid = (k < Sn);
  float en = 0.f, v = -3.4e38f;
  if (valid) {
    int j = q * Sn + k;
    float e = E[(size_t)bh * SS + j];
    en = ((e - mu2[j]) * rsqrtf(var2[j] + EPS) * g2[j] + b2[j]) * (1.f / SQRT_C);
    v = en;
  }
  for (int off = 16; off > 0; off >>= 1) v = fmaxf(v, __shfl_xor(v, off));
  float ex = valid ? __expf(en - v) : 0.f;
  float s = ex;
  for (int off = 16; off > 0; off >>= 1) s += __shfl_xor(s, off);
  if (valid) ATT[(size_t)bh * SS + q * Sn + k] = ex / s;
}

// O[b,q,h,dh] = sum_l ATT[bh,q,l]*V[b,l,h,dh]; K padded 31->32.
// grid = BH, block = 256 (8 waves: 2 M-tiles x 4 N-tiles).
__global__ void self_av(const float* __restrict__ ATT, const float* __restrict__ V,
                        float* __restrict__ O) {
  const int bh = blockIdx.x, b = bh >> 3, h = bh & 7;
  const int w = threadIdx.x >> 5, lane = threadIdx.x & 31;
  const int mt = w >> 2, nt = w & 3;
  const int lm = lane & 15, ko = (lane >> 4) * 2;
  const float* ap = ATT + (size_t)bh * SS;
  const float* vp = V + (size_t)b * Sn * Cn + h * 64;
  const int mr = mt * 16 + lm;   // query row
  const int nc = nt * 16 + lm;   // dh column
  v8f acc = {};
#pragma unroll
  for (int kk = 0; kk < 32; kk += 4) {
    int c0 = kk + ko, c1 = c0 + 1;
    v2f a, bf;
    a.x = (mr < Sn && c0 < Sn) ? ap[(size_t)mr * Sn + c0] : 0.f;
    a.y = (mr < Sn && c1 < Sn) ? ap[(size_t)mr * Sn + c1] : 0.f;
    bf.x = (c0 < Sn) ? vp[(size_t)c0 * Cn + nc] : 0.f;
    bf.y = (c1 < Sn) ? vp[(size_t)c1 * Cn + nc] : 0.f;
    acc = WMMA_F32(a, bf, acc);
  }
#pragma unroll
  for (int r = 0; r < 8; ++r) {
    int m = mt * 16 + r + (lane >> 4) * 8;
    int n = nt * 16 + lm;
    if (m < Sn) O[((size_t)b * Sn + m) * Cn + h * 64 + n] = acc[r];
  }
}

// Row LayerNorm (+ optional SiLU). grid = rows, block = 256.
__global__ void layernorm_rows(const float* __restrict__ x, const float* __restrict__ g,
                               const float* __restrict__ bta, float* __restrict__ y,
                               int C, int do_silu) {
  const int row = blockIdx.x;
  const float* xr = x + (size_t)row * C;
  float* yr = y + (size_t)row * C;
  __shared__ float s1[256], s2[256];
  float sum = 0.f, sq = 0.f;
  for (int i = threadIdx.x; i < C; i += 256) { float v = xr[i]; sum += v; sq += v * v; }
  s1[threadIdx.x] = sum; s2[threadIdx.x] = sq;
  __syncthreads();
  for (int s = 128; s > 0; s >>= 1) {
    if (threadIdx.x < s) { s1[threadIdx.x] += s1[threadIdx.x + s];
                           s2[threadIdx.x] += s2[threadIdx.x + s]; }
    __syncthreads();
  }
  float mu = s1[0] / C;
  float inv = rsqrtf(s2[0] / C - mu * mu + EPS);
  for (int i = threadIdx.x; i < C; i += 256) {
    float v = (xr[i] - mu) * inv * g[i] + bta[i];
    if (do_silu) v = v / (1.f + __expf(-v));
    yr[i] = v;
  }
}

// Cross energy: E2[bh,q,n] = sum_dh Q2[b,q,h,dh]*K2[b,n,h,dh]  (raw; output #2)
// grid = (CTX/64, BH), block = 128; each wave: 2 M-tiles share each B frag.
__global__ void cross_energy(const float* __restrict__ Q2, const float* __restrict__ K2,
                             float* __restrict__ E2) {
  const int bh = blockIdx.y, b = bh >> 3, h = bh & 7;
  const int w = threadIdx.x >> 5, lane = threadIdx.x & 31;
  const int n0 = blockIdx.x * 64 + w * 16;
  const int lm = lane & 15, ko = (lane >> 4) * 2;
  const float* qp = Q2 + (size_t)b * Sn * Cn + h * 64;
  const float* kp = K2 + (size_t)b * CTXn * Cn + h * 64;
  const int nr = n0 + lm;
  const int mr0 = lm;          // tile 0 rows 0..15 (all valid)
  const int mr1 = 16 + lm;     // tile 1 rows 16..31 (31 invalid)
  v8f acc0 = {}, acc1 = {};
#pragma unroll
  for (int kk = 0; kk < 64; kk += 4) {
    v2f bf = *(const v2f*)(kp + (size_t)nr * Cn + kk + ko);
    v2f a0 = *(const v2f*)(qp + (size_t)mr0 * Cn + kk + ko);
    v2f a1;
    if (mr1 < Sn) a1 = *(const v2f*)(qp + (size_t)mr1 * Cn + kk + ko);
    else { a1.x = 0.f; a1.y = 0.f; }
    acc0 = WMMA_F32(a0, bf, acc0);
    acc1 = WMMA_F32(a1, bf, acc1);
  }
#pragma unroll
  for (int r = 0; r < 8; ++r) {
    int mlo = r + (lane >> 4) * 8;
    int n = n0 + lm;
    E2[((size_t)bh * Sn + mlo) * CTXn + n] = acc0[r];
    int mhi = 16 + mlo;
    if (mhi < Sn) E2[((size_t)bh * Sn + mhi) * CTXn + n] = acc1[r];
  }
}

// softmax over CTX of E2/sqrt(C). grid = BH*S rows, block = 256.
__global__ void softmax_cross(const float* __restrict__ E2, float* __restrict__ A2) {
  const int row = blockIdx.x;
  const float* e = E2 + (size_t)row * CTXn;
  float* a = A2 + (size_t)row * CTXn;
  __shared__ float red[256];
  float m = -3.4e38f;
  for (int i = threadIdx.x; i < CTXn; i += 256) m = fmaxf(m, e[i] * (1.f / SQRT_C));
  red[threadIdx.x] = m; __syncthreads();
  for (int s = 128; s > 0; s >>= 1) {
    if (threadIdx.x < s) red[threadIdx.x] = fmaxf(red[threadIdx.x], red[threadIdx.x + s]);
    __syncthreads();
  }
  m = red[0];
  __syncthreads();
  float sum = 0.f;
  for (int i = threadIdx.x; i < CTXn; i += 256) {
    float ex = __expf(e[i] * (1.f / SQRT_C) - m);
    a[i] = ex; sum += ex;
  }
  red[threadIdx.x] = sum; __syncthreads();
  for (int s = 128; s > 0; s >>= 1) {
    if (threadIdx.x < s) red[threadIdx.x] += red[threadIdx.x + s];
    __syncthreads();
  }
  float inv = 1.f / red[0];
  for (int i = threadIdx.x; i < CTXn; i += 256) a[i] *= inv;
}

// O2[b,q,h,dh] = sum_l A2[bh,q,l]*V2[b,l,h,dh]; K=2048.
// grid = BH, block = 256 (2 M-tiles x 4 N-tiles). Double-buffered like gemm.
__global__ void __launch_bounds__(256)
cross_av(const float* __restrict__ A2, const float* __restrict__ V2,
         float* __restrict__ O2) {
  const int bh = blockIdx.x, b = bh >> 3, h = bh & 7;
  const int w = threadIdx.x >> 5, lane = threadIdx.x & 31;
  const int mt = w >> 2, nt = w & 3;
  const int lm = lane & 15, ko = (lane >> 4) * 2;
  const float* ap = A2 + (size_t)bh * Sn * CTXn;
  const float* vp = V2 + (size_t)b * CTXn * Cn + h * 64;
  const int mr = mt * 16 + lm;
  const int nc = nt * 16 + lm;
  const bool mok = (mr < Sn);
  const float* arow = ap + (size_t)mr * CTXn + ko;   // only read when mok
  v8f acc = {};
  v2f a, bf;
  if (mok) a = *(const v2f*)(arow);
  else { a.x = 0.f; a.y = 0.f; }
  bf.x = vp[(size_t)(0 + ko) * Cn + nc];
  bf.y = vp[(size_t)(1 + ko) * Cn + nc];
#pragma unroll 2
  for (int kk = 4; kk < CTXn; kk += 4) {
    v2f an, bn;
    if (mok) an = *(const v2f*)(arow + kk);
    else { an.x = 0.f; an.y = 0.f; }
    bn.x = vp[(size_t)(kk + ko) * Cn + nc];
    bn.y = vp[(size_t)(kk + ko + 1) * Cn + nc];
    acc = WMMA_F32(a, bf, acc);
    a = an; bf = bn;
  }
  acc = WMMA_F32(a, bf, acc);
#pragma unroll
  for (int r = 0; r < 8; ++r) {
    int m = mt * 16 + r + (lane >> 4) * 8;
    int n = nt * 16 + lm;
    if (m < Sn) O2[((size_t)b * Sn + m) * Cn + h * 64 + n] = acc[r];
  }
}

// ============================================================
extern "C" void kernel_launch(void* const* d_in, const int* in_sizes, int n_in,
                              void* d_out, int out_size, void* d_ws, size_t ws_size,
                              hipStream_t stream) {
  const float* queries  = (const float*)d_in[0];
  const float* contexts = (const float*)d_in[1];
  const float* sa_bn_g  = (const float*)d_in[2];
  const float* sa_bn_b  = (const float*)d_in[3];
  const float* sa_q_w   = (const float*)d_in[4];
  const float* sa_k_w   = (const float*)d_in[5];
  const float* sa_v_w   = (const float*)d_in[6];
  const float* sa_out_w = (const float*)d_in[7];
  const float* sa_out_b = (const float*)d_in[8];
  const float* sa_pre_g = (const float*)d_in[9];
  const float* sa_pre_b = (const float*)d_in[10];
  const float* ca_ln_g  = (const float*)d_in[11];
  const float* ca_ln_b  = (const float*)d_in[12];
  const float* ca_q_w   = (const float*)d_in[13];
  const float* ca_k_w   = (const float*)d_in[14];
  const float* ca_v_w   = (const float*)d_in[15];
  const float* ca_out_w = (const float*)d_in[16];
  const float* ca_out_b = (const float*)d_in[17];
  const float* ff_ln1_g = (const float*)d_in[18];
  const float* ff_ln1_b = (const float*)d_in[19];
  const float* ff_w1    = (const float*)d_in[20];
  const float* ff_ln2_g = (const float*)d_in[21];
  const float* ff_ln2_b = (const float*)d_in[22];
  const float* ff_w2    = (const float*)d_in[23];

  float* ws = (float*)d_ws;
  float* out1 = (float*)d_out;                    // (B,S,C)
  float* out2 = out1 + NQ * Cn;                   // energy2 (B,H,S,CTX)

  float* MU1 = ws + OFF_MU1; float* VAR1 = ws + OFF_VAR1;
  float* MU2 = ws + OFF_MU2; float* VAR2 = ws + OFF_VAR2;
  float* XN  = ws + OFF_XN;  float* Qb = ws + OFF_Q;  float* Kb = ws + OFF_K;
  float* Vb  = ws + OFF_V;   float* EN = ws + OFF_EN; float* ATT = ws + OFF_ATT;
  float* Ob  = ws + OFF_O;   float* XSA = ws + OFF_XSA; float* QN = ws + OFF_QN;
  float* Q2  = ws + OFF_Q2;  float* O2 = ws + OFF_O2;  float* XCA = ws + OFF_XCA;
  float* H1  = ws + OFF_H1;  float* H1N = ws + OFF_H1N;
  float* K2  = ws + OFF_K2;  float* V2 = ws + OFF_V2;  float* A2 = ws + OFF_A2;

  const float* NUL = nullptr;

  // ---- Self attention ----
  bn_stats_ch<<<Cn, 256, 0, stream>>>(queries, (int)NQ, Cn, MU1, VAR1);
  bn_apply<<<(int)((NQ * Cn + 255) / 256), 256, 0, stream>>>(
      queries, MU1, VAR1, sa_bn_g, sa_bn_b, XN, (int)(NQ * Cn), Cn);
  dim3 gQ(Cn / 64, (unsigned)(NQ / 64));
  gemm_xwt<<<gQ, 128, 0, stream>>>(XN, sa_q_w, NUL, NUL, Qb, (int)NQ, Cn, Cn);
  gemm_xwt<<<gQ, 128, 0, stream>>>(XN, sa_k_w, NUL, NUL, Kb, (int)NQ, Cn, Cn);
  gemm_xwt<<<gQ, 128, 0, stream>>>(XN, sa_v_w, NUL, NUL, Vb, (int)NQ, Cn, Cn);
  self_energy<<<(int)BH, 128, 0, stream>>>(Qb, Kb, EN);
  bn_stats_ch<<<(int)SS, 256, 0, stream>>>(EN, (int)BH, (int)SS, MU2, VAR2);
  softmax_self<<<(int)(BH * Sn), 32, 0, stream>>>(EN, MU2, VAR2, sa_pre_g, sa_pre_b, ATT);
  self_av<<<(int)BH, 256, 0, stream>>>(ATT, Vb, Ob);
  gemm_xwt<<<gQ, 128, 0, stream>>>(Ob, sa_out_w, sa_out_b, queries, XSA, (int)NQ, Cn, Cn);

  // ---- Cross attention ----
  layernorm_rows<<<(int)NQ, 256, 0, stream>>>(XSA, ca_ln_g, ca_ln_b, QN, Cn, 0);
  gemm_xwt<<<gQ, 128, 0, stream>>>(QN, ca_q_w, NUL, NUL, Q2, (int)NQ, Cn, Cn);
  dim3 gKV(Cn / 64, (unsigned)(NCTX / 64));
  gemm_xwt<<<gKV, 128, 0, stream>>>(contexts, ca_k_w, NUL, NUL, K2, (int)NCTX, Cn, Cn);
  gemm_xwt<<<gKV, 128, 0, stream>>>(contexts, ca_v_w, NUL, NUL, V2, (int)NCTX, Cn, Cn);
  dim3 gE2(CTXn / 64, (unsigned)BH);
  cross_energy<<<gE2, 128, 0, stream>>>(Q2, K2, out2);
  softmax_cross<<<(int)(BH * Sn), 256, 0, stream>>>(out2, A2);   // A2 reuses K2 region
  cross_av<<<(int)BH, 256, 0, stream>>>(A2, V2, O2);
  gemm_xwt<<<gQ, 128, 0, stream>>>(O2, ca_out_w, ca_out_b, XSA, XCA, (int)NQ, Cn, Cn);

  // ---- FFN ----
  layernorm_rows<<<(int)NQ, 256, 0, stream>>>(XCA, ff_ln1_g, ff_ln1_b, QN, Cn, 1);
  dim3 gH1(2 * Cn / 64, (unsigned)(NQ / 64));
  gemm_xwt<<<gH1, 128, 0, stream>>>(QN, ff_w1, NUL, NUL, H1, (int)NQ, 2 * Cn, Cn);
  layernorm_rows<<<(int)NQ, 256, 0, stream>>>(H1, ff_ln2_g, ff_ln2_b, H1N, 2 * Cn, 1);
  gemm_xwt<<<gQ, 128, 0, stream>>>(H1N, ff_w2, NUL, XCA, out1, (int)NQ, Cn, 2 * Cn);
}